// GCN_14370960573165
// MI455X (gfx1250) — compile-verified
//
#include <hip/hip_runtime.h>
#include <hip/hip_bf16.h>

typedef __attribute__((ext_vector_type(2))) float v2f;
typedef __attribute__((ext_vector_type(8))) float v8f;

#define HDIM 128
#define NGRAPH 64
#define NCLASS 4

// ---------- degree / norm ----------
__global__ void k_fill1(float* __restrict__ p, int n) {
    int i = blockIdx.x * blockDim.x + threadIdx.x;
    if (i < n) p[i] = 1.0f;   // self-loop contributes 1 to every degree
}

__global__ void k_deg(const int* __restrict__ dst, float* __restrict__ deg, int E) {
    int i = blockIdx.x * blockDim.x + threadIdx.x;
    if (i < E) atomicAdd(&deg[dst[i]], 1.0f);
}

__global__ void k_rsqrt(float* __restrict__ p, int n) {
    int i = blockIdx.x * blockDim.x + threadIdx.x;
    if (i < n) p[i] = rsqrtf(p[i]);   // deg >= 1 always (self-loops)
}

// ---------- layer 1: g = dinv * (x @ W1), x is [N,3] ----------
__global__ void k_layer1(const float* __restrict__ x, const float* __restrict__ W1,
                         const float* __restrict__ dinv, float* __restrict__ G, int N) {
    int t = blockIdx.x * blockDim.x + threadIdx.x;
    if (t >= N * HDIM) return;
    int i = t >> 7, j = t & (HDIM - 1);
    float h = x[i * 3 + 0] * W1[0 * HDIM + j]
            + x[i * 3 + 1] * W1[1 * HDIM + j]
            + x[i * 3 + 2] * W1[2 * HDIM + j];
    G[t] = dinv[i] * h;
}

// ---------- edge scatter: ACC[dst] += G[src]; one wave per edge, float4/lane ----------
__global__ void k_scatter(const int* __restrict__ src, const int* __restrict__ dst,
                          const float* __restrict__ G, float* __restrict__ ACC, int E) {
    long long t = (long long)blockIdx.x * blockDim.x + threadIdx.x;
    if (t >= (long long)E * 32) return;
    int e = (int)(t >> 5);
    int c = ((int)t & 31) << 2;             // feature offset 0..124
    int s = src[e], d = dst[e];
    const float4 v = *(const float4*)(G + (size_t)s * HDIM + c);
    float* o = ACC + (size_t)d * HDIM + c;
    atomicAdd(o + 0, v.x); atomicAdd(o + 1, v.y);
    atomicAdd(o + 2, v.z); atomicAdd(o + 3, v.w);
}

// ---------- epilogue: ACC = act(dinv*(ACC + G) + b)  (self-loop folded in) ----------
__global__ void k_finalize(float* __restrict__ ACC, const float* __restrict__ G,
                           const float* __restrict__ dinv, const float* __restrict__ b,
                           int N, int do_relu) {
    int t = blockIdx.x * blockDim.x + threadIdx.x;
    if (t >= N * HDIM) return;
    int i = t >> 7, j = t & (HDIM - 1);
    float v = dinv[i] * (ACC[t] + G[t]) + b[j];
    if (do_relu) v = fmaxf(v, 0.0f);
    ACC[t] = v;
}

// ---------- WMMA GEMM: G = dinv * (Hin @ W), Hin [N,128], W [128,128] ----------
// One wave per 16-row tile; V_WMMA_F32_16X16X4_F32, 32 K-steps, 8 column tiles.
__global__ void k_gemm_wmma(const float* __restrict__ Hin, const float* __restrict__ W,
                            const float* __restrict__ dinv, float* __restrict__ G, int N) {
    int wave = blockIdx.x * (blockDim.x >> 5) + (threadIdx.x >> 5);
    int lane = threadIdx.x & 31;
    int row0 = wave << 4;
    if (row0 >= N) return;                  // wave-uniform: EXEC stays all-1s below
    int half = lane >> 4;                   // 0: lanes 0-15, 1: lanes 16-31
    int l = lane & 15;
    int arow = row0 + l; if (arow >= N) arow = N - 1;   // clamp loads for partial tiles
    const float* hrow = Hin + (size_t)arow * HDIM;

    for (int nt = 0; nt < 8; ++nt) {
        int col0 = nt << 4;
        v8f acc = {};
#pragma unroll
        for (int k = 0; k < HDIM; k += 4) {
            int kk = k + (half << 1);
            v2f a, b;
            a[0] = hrow[kk];                                  // A: 16x4 tile, ISA layout
            a[1] = hrow[kk + 1];
            b[0] = W[(size_t)kk * HDIM + col0 + l];           // B: 4x16 tile, ISA layout
            b[1] = W[(size_t)(kk + 1) * HDIM + col0 + l];
            acc = __builtin_amdgcn_wmma_f32_16x16x4_f32(
                false, a, false, b, (short)0, acc, false, false);
        }
#pragma unroll
        for (int r = 0; r < 8; ++r) {                         // C/D: VGPR r -> M = r + 8*half
            int row = row0 + r + (half << 3);
            if (row < N) G[(size_t)row * HDIM + col0 + l] = dinv[row] * acc[r];
        }
    }
}

// ---------- mean pool ----------
__global__ void k_pool(const float* __restrict__ Hf, const int* __restrict__ batch,
                       float* __restrict__ sums, float* __restrict__ cnts, int N) {
    long long t = (long long)blockIdx.x * blockDim.x + threadIdx.x;
    if (t >= (long long)N * 32) return;
    int i = (int)(t >> 5);
    int c = ((int)t & 31) << 2;
    int g = batch[i];
    const float4 v = *(const float4*)(Hf + (size_t)i * HDIM + c);
    float* o = sums + (size_t)g * HDIM + c;
    atomicAdd(o + 0, v.x); atomicAdd(o + 1, v.y);
    atomicAdd(o + 2, v.z); atomicAdd(o + 3, v.w);
    if (c == 0) atomicAdd(&cnts[g], 1.0f);
}

// ---------- head: out[g][c] = (sums[g]/cnt[g]) @ Wlin + blin ----------
__global__ void k_head(const float* __restrict__ sums, const float* __restrict__ cnts,
                       const float* __restrict__ Wlin, const float* __restrict__ blin,
                       float* __restrict__ out) {
    int t = threadIdx.x;                    // one block of 256 = 64 graphs x 4 classes
    int g = t >> 2, c = t & 3;
    float inv = 1.0f / fmaxf(cnts[g], 1.0f);
    float acc = blin[c];
    for (int j = 0; j < HDIM; ++j)
        acc += (sums[g * HDIM + j] * inv) * Wlin[j * NCLASS + c];
    out[g * NCLASS + c] = acc;
}

extern "C" void kernel_launch(void* const* d_in, const int* in_sizes, int n_in,
                              void* d_out, int out_size, void* d_ws, size_t ws_size,
                              hipStream_t stream) {
    const float* x    = (const float*)d_in[0];
    const int*   eidx = (const int*)d_in[1];
    const int*   batch= (const int*)d_in[2];
    const float* W1   = (const float*)d_in[3];
    const float* b1   = (const float*)d_in[4];
    const float* W2   = (const float*)d_in[5];
    const float* b2   = (const float*)d_in[6];
    const float* W3   = (const float*)d_in[7];
    const float* b3   = (const float*)d_in[8];
    const float* Wlin = (const float*)d_in[9];
    const float* blin = (const float*)d_in[10];
    (void)n_in; (void)out_size; (void)ws_size;

    const int N = in_sizes[0] / 3;
    const int E = in_sizes[1] / 2;
    const int* src = eidx;
    const int* dst = eidx + E;

    // workspace layout (floats): dinv | G (N*128) | ACC (N*128) | sums (64*128) | cnts (64)
    float* ws   = (float*)d_ws;
    size_t NH   = (size_t)N * HDIM;
    float* dinv = ws;
    float* G    = ws + (((size_t)N + 255) & ~(size_t)255);
    float* ACC  = G + NH;
    float* sums = ACC + NH;
    float* cnts = sums + (size_t)NGRAPH * HDIM;

    const int TPB = 256;
    dim3 blkN((N + TPB - 1) / TPB);
    dim3 blkE((E + TPB - 1) / TPB);
    dim3 blkNH(((int)NH + TPB - 1) / TPB);
    long long scat = (long long)E * 32;
    dim3 blkScat((unsigned)((scat + TPB - 1) / TPB));
    long long poolT = (long long)N * 32;
    dim3 blkPool((unsigned)((poolT + TPB - 1) / TPB));
    int rowTiles = (N + 15) / 16;
    dim3 blkGemm((rowTiles + 7) / 8);       // 8 waves (256 threads) per block

    // degree -> dinv = (deg)^{-1/2}
    k_fill1<<<blkN, TPB, 0, stream>>>(dinv, N);
    k_deg  <<<blkE, TPB, 0, stream>>>(dst, dinv, E);
    k_rsqrt<<<blkN, TPB, 0, stream>>>(dinv, N);

    // layer 1
    k_layer1<<<blkNH, TPB, 0, stream>>>(x, W1, dinv, G, N);
    hipMemsetAsync(ACC, 0, NH * sizeof(float), stream);
    k_scatter <<<blkScat, TPB, 0, stream>>>(src, dst, G, ACC, E);
    k_finalize<<<blkNH, TPB, 0, stream>>>(ACC, G, dinv, b1, N, 1);   // ACC = h1

    // layer 2
    k_gemm_wmma<<<blkGemm, TPB, 0, stream>>>(ACC, W2, dinv, G, N);   // G = dinv*(h1@W2)
    hipMemsetAsync(ACC, 0, NH * sizeof(float), stream);
    k_scatter <<<blkScat, TPB, 0, stream>>>(src, dst, G, ACC, E);
    k_finalize<<<blkNH, TPB, 0, stream>>>(ACC, G, dinv, b2, N, 1);   // ACC = h2

    // layer 3 (no relu)
    k_gemm_wmma<<<blkGemm, TPB, 0, stream>>>(ACC, W3, dinv, G, N);   // G = dinv*(h2@W3)
    hipMemsetAsync(ACC, 0, NH * sizeof(float), stream);
    k_scatter <<<blkScat, TPB, 0, stream>>>(src, dst, G, ACC, E);
    k_finalize<<<blkNH, TPB, 0, stream>>>(ACC, G, dinv, b3, N, 0);   // ACC = h3

    // mean pool + head
    hipMemsetAsync(sums, 0, (size_t)(NGRAPH * HDIM + NGRAPH) * sizeof(float), stream);
    k_pool<<<blkPool, TPB, 0, stream>>>(ACC, batch, sums, cnts, N);
    k_head<<<1, 256, 0, stream>>>(sums, cnts, Wlin, blin, (float*)d_out);
}